// PointNetSetAbstraction_40381282517460
// MI455X (gfx1250) — compile-verified
//
#include <hip/hip_runtime.h>

#define B_    16
#define N_    4096
#define CIN0  64
#define S_    1024
#define K_    32
#define M_    (B_ * S_ * K_)     // 524288 rows through the MLP
#define R2_   0.16f
#define EPS_  1e-5f
#define ITERS 8

typedef __attribute__((ext_vector_type(2))) float v2f;
typedef __attribute__((ext_vector_type(8))) float v8f;

// ---------------------------------------------------------------------------
// 1) Farthest point sampling: one workgroup per batch, xyz + distance in LDS.
//    Matches reference: start index 0, min-distance update, first-max argmax.
// ---------------------------------------------------------------------------
__global__ __launch_bounds__(256) void fps_kernel(const float* __restrict__ xyz,
                                                  float* __restrict__ new_xyz) {
  extern __shared__ float smem[];
  float* xs   = smem;
  float* ys   = xs + N_;
  float* zs   = ys + N_;
  float* dist = zs + N_;
  float* wv   = dist + N_;          // 8 per-wave max values
  int*   wi   = (int*)(wv + 8);     // 8 per-wave argmax indices
  int*   sFar = wi + 8;             // broadcast slot

  const int b = blockIdx.x, tid = threadIdx.x;
  const int lane = tid & 31, wid = tid >> 5;

  for (int i = tid; i < N_; i += 256) {
    const float* p = xyz + ((size_t)b * N_ + i) * 3;
    xs[i] = p[0]; ys[i] = p[1]; zs[i] = p[2];
    dist[i] = 1e10f;
  }
  __syncthreads();

  int far = 0;
  for (int s = 0; s < S_; ++s) {
    if (tid == 0) {
      float* o = new_xyz + ((size_t)b * S_ + s) * 3;
      o[0] = xs[far]; o[1] = ys[far]; o[2] = zs[far];
    }
    const float cx = xs[far], cy = ys[far], cz = zs[far];
    float bv = -1.0f; int bi = 0;
#pragma unroll
    for (int j = 0; j < N_ / 256; ++j) {
      const int i = tid + j * 256;
      const float dx = xs[i] - cx, dy = ys[i] - cy, dz = zs[i] - cz;
      const float d  = dx * dx + dy * dy + dz * dz;
      const float nd = fminf(dist[i], d);
      dist[i] = nd;
      if (nd > bv) { bv = nd; bi = i; }   // ascending i => keeps first max
    }
    // wave32 argmax reduce, first-index tie-break
    for (int off = 16; off > 0; off >>= 1) {
      const float ov = __shfl_xor(bv, off);
      const int   oi = __shfl_xor(bi, off);
      if (ov > bv || (ov == bv && oi < bi)) { bv = ov; bi = oi; }
    }
    if (lane == 0) { wv[wid] = bv; wi[wid] = bi; }
    __syncthreads();
    if (tid == 0) {
      float mv = wv[0]; int mi = wi[0];
      for (int w = 1; w < 8; ++w)
        if (wv[w] > mv || (wv[w] == mv && wi[w] < mi)) { mv = wv[w]; mi = wi[w]; }
      *sFar = mi;
    }
    __syncthreads();
    far = *sFar;
  }
}

// ---------------------------------------------------------------------------
// 2) Ball query + grouping: one wave per centroid (8 centroids / block).
//    Ballot rank-compaction selects the first <=32 in-ball indices in order,
//    pads with the first neighbor; writes [M, 68] padded feature rows.
// ---------------------------------------------------------------------------
__global__ __launch_bounds__(256) void group_kernel(const float* __restrict__ xyz,
                                                    const float* __restrict__ points,
                                                    const float* __restrict__ new_xyz,
                                                    float* __restrict__ feat) {
  __shared__ int sel[8][K_];
  const int tid = threadIdx.x, lane = tid & 31, w = tid >> 5;
  const int bs = blockIdx.x * 8 + w;        // = b*1024 + s
  const int b  = bs >> 10;

  const float* c = new_xyz + (size_t)bs * 3;
  const float cx = c[0], cy = c[1], cz = c[2];

  int cnt = 0;
  for (int base = 0; base < N_ && cnt < K_; base += 32) {
    const int i = base + lane;
    const float* p = xyz + ((size_t)b * N_ + i) * 3;
    const float dx = p[0] - cx, dy = p[1] - cy, dz = p[2] - cz;
    const float sq = dx * dx + dy * dy + dz * dz;
    const bool inb = (sq <= R2_);
    const unsigned mask = (unsigned)__ballot(inb);
    const int rank = __popc(mask & ((1u << lane) - 1u));
    if (inb) {
      const int slot = cnt + rank;
      if (slot < K_) sel[w][slot] = i;
    }
    cnt += __popc(mask);
  }
  if (cnt > K_) cnt = K_;
  if (cnt == 0) { sel[w][0] = N_ - 1; cnt = 1; }  // OOB-gather clamp semantics

  const int gi = sel[w][lane < cnt ? lane : 0];   // pad with first neighbor

  const size_t row = (size_t)bs * K_ + lane;
  float* fr = feat + row * 68;
  const float* p  = xyz    + ((size_t)b * N_ + gi) * 3;
  const float* pt = points + ((size_t)b * N_ + gi) * CIN0;
  fr[0] = p[0] - cx; fr[1] = p[1] - cy; fr[2] = p[2] - cz;
#pragma unroll
  for (int cc = 0; cc < CIN0; ++cc) fr[3 + cc] = pt[cc];
  fr[67] = 0.0f;                                   // K-dim pad (W pad row is 0 too)
}

// ---------------------------------------------------------------------------
// 3) 1x1-conv layers as WMMA fp32 GEMM: D[16x16] per wave via
//    V_WMMA_F32_16X16X4_F32.  Optionally applies previous layer's BN+ReLU on
//    the A operand while loading.  Fused per-channel sum / sum-of-squares for
//    this layer's BN (LDS ds_add_f32 then one global atomic per channel).
// ---------------------------------------------------------------------------
template <int COUT, int CIN, int CSRC, bool PRE>
__global__ __launch_bounds__(256) void mlp_wmma_kernel(
    const float* __restrict__ X, const float* __restrict__ W,
    const float* __restrict__ bias,
    const float* __restrict__ preScale, const float* __restrict__ preShift,
    float* __restrict__ Y, float* __restrict__ gSum, float* __restrict__ gSq) {
  constexpr int NT = COUT / 16;   // N tiles per block row
  constexpr int MT = 8 / NT;      // M tiles covered by the 8 waves
  __shared__ float Wl[COUT * CIN];
  __shared__ float biasL[COUT];
  __shared__ float scL[CIN], shL[CIN];
  __shared__ float bSum[COUT], bSq[COUT];

  const int tid = threadIdx.x;
  for (int idx = tid; idx < COUT * CIN; idx += 256) {
    const int o = idx / CIN, cc = idx % CIN;
    Wl[idx] = (cc < CSRC) ? W[o * CSRC + cc] : 0.0f;  // zero-pad K dim
  }
  for (int idx = tid; idx < COUT; idx += 256) {
    biasL[idx] = bias[idx]; bSum[idx] = 0.0f; bSq[idx] = 0.0f;
  }
  for (int idx = tid; idx < CIN; idx += 256) {
    if (PRE) { scL[idx] = preScale[idx]; shL[idx] = preShift[idx]; }
    else     { scL[idx] = 1.0f;          shL[idx] = 0.0f; }
  }
  __syncthreads();

  const int lane  = tid & 31, w = tid >> 5;
  const int ntile = w % NT,   msub = w / NT;
  const int nn    = ntile * 16 + (lane & 15);  // output channel of this lane
  const int kb    = (lane >> 4) * 2;           // A/B frag k-pair select
  const int r     = lane & 15;                 // A frag row
  const int rbase = (lane >> 4) * 8;           // C/D frag row base

  for (int it = 0; it < ITERS; ++it) {
    const size_t mt   = ((size_t)blockIdx.x * ITERS + it) * MT + msub;
    const size_t row0 = mt * 16;
    const float* xr   = X + (row0 + r) * CIN;
    v8f acc = {0.f, 0.f, 0.f, 0.f, 0.f, 0.f, 0.f, 0.f};
#pragma unroll
    for (int k0 = 0; k0 < CIN; k0 += 4) {
      float a0 = xr[k0 + kb], a1 = xr[k0 + kb + 1];
      if (PRE) {  // previous layer's BN + ReLU applied on the fly
        a0 = fmaxf(a0 * scL[k0 + kb]     + shL[k0 + kb],     0.0f);
        a1 = fmaxf(a1 * scL[k0 + kb + 1] + shL[k0 + kb + 1], 0.0f);
      }
      v2f a;  a.x = a0;  a.y = a1;
      v2f bb; bb.x = Wl[nn * CIN + k0 + kb];
              bb.y = Wl[nn * CIN + k0 + kb + 1];
      acc = __builtin_amdgcn_wmma_f32_16x16x4_f32(
          /*neg_a=*/false, a, /*neg_b=*/false, bb,
          /*c_mod=*/(short)0, acc, /*reuse_a=*/false, /*reuse_b=*/false);
    }
    const float bo = biasL[nn];
    float cs = 0.0f, cq = 0.0f;
    float* yb = Y + (row0 + rbase) * COUT + nn;
#pragma unroll
    for (int v = 0; v < 8; ++v) {
      const float y = acc[v] + bo;          // bias before BN stats (as reference)
      yb[(size_t)v * COUT] = y;
      cs += y; cq += y * y;
    }
    cs += __shfl_xor(cs, 16);               // fold rows 8..15 half of the wave
    cq += __shfl_xor(cq, 16);
    if (lane < 16) { atomicAdd(&bSum[nn], cs); atomicAdd(&bSq[nn], cq); }
  }
  __syncthreads();
  for (int idx = tid; idx < COUT; idx += 256) {
    atomicAdd(&gSum[idx], bSum[idx]);
    atomicAdd(&gSq[idx], bSq[idx]);
  }
}

// ---------------------------------------------------------------------------
// 4) Turn channel sums into BN scale/shift:  y_norm = scale*y + shift
// ---------------------------------------------------------------------------
__global__ void bn_prep_kernel(const float* __restrict__ sum, const float* __restrict__ sq,
                               const float* __restrict__ g, const float* __restrict__ be,
                               float* __restrict__ scale, float* __restrict__ shift, int C) {
  const int o = threadIdx.x;
  if (o < C) {
    const float inv = 1.0f / (float)M_;
    const float m = sum[o] * inv;
    const float v = sq[o] * inv - m * m;     // biased variance
    const float a = g[o] * rsqrtf(v + EPS_);
    scale[o] = a;
    shift[o] = be[o] - m * a;
  }
}

// ---------------------------------------------------------------------------
// 5) Final BN + ReLU + max over the K=32 samples -> new_points [B,S,128]
// ---------------------------------------------------------------------------
__global__ __launch_bounds__(128) void maxpool_kernel(const float* __restrict__ act,
                                                      const float* __restrict__ scale,
                                                      const float* __restrict__ shift,
                                                      float* __restrict__ outp) {
  const int bs = blockIdx.x, o = threadIdx.x;
  const float a = scale[o], sh = shift[o];
  const float* base = act + ((size_t)bs * K_) * 128 + o;
  float m = 0.0f;                                // relu outputs are >= 0
#pragma unroll
  for (int k = 0; k < K_; ++k) {
    const float y = fmaxf(base[(size_t)k * 128] * a + sh, 0.0f);
    m = fmaxf(m, y);
  }
  outp[(size_t)bs * 128 + o] = m;
}

__global__ void zero_kernel(float* p, int n) {
  const int i = blockIdx.x * 256 + threadIdx.x;
  if (i < n) p[i] = 0.0f;
}

// ---------------------------------------------------------------------------
extern "C" void kernel_launch(void* const* d_in, const int* in_sizes, int n_in,
                              void* d_out, int out_size, void* d_ws, size_t ws_size,
                              hipStream_t stream) {
  (void)in_sizes; (void)n_in; (void)out_size; (void)ws_size;
  const float* xyz    = (const float*)d_in[0];
  const float* points = (const float*)d_in[1];
  const float* w0 = (const float*)d_in[2];
  const float* b0 = (const float*)d_in[3];
  const float* g0 = (const float*)d_in[4];
  const float* be0 = (const float*)d_in[5];
  const float* w1 = (const float*)d_in[6];
  const float* b1 = (const float*)d_in[7];
  const float* g1 = (const float*)d_in[8];
  const float* be1 = (const float*)d_in[9];
  const float* w2 = (const float*)d_in[10];
  const float* b2 = (const float*)d_in[11];
  const float* g2 = (const float*)d_in[12];
  const float* be2 = (const float*)d_in[13];

  float* out        = (float*)d_out;
  float* new_xyz    = out;                          // [B,S,3]
  float* new_points = out + (size_t)B_ * S_ * 3;    // [B,S,128]

  // workspace carve-up
  char*  ws    = (char*)d_ws;
  float* stats = (float*)ws;            // 3 x (sum[128], sq[128]) = 768 floats
  float* scsh  = stats + 768;           // 3 x (scale[128], shift[128])
  float* feat  = (float*)(ws + 8192);                 // [M,68]  (acts as act2 later)
  float* act1  = feat + (size_t)M_ * 68;              // [M,64]
  float* act3  = act1 + (size_t)M_ * 64;              // [M,128]
  float* act2  = feat;                                // reuse: feat dead after layer 1

  float* sum0 = stats;       float* sq0 = stats + 128;
  float* sum1 = stats + 256; float* sq1 = stats + 384;
  float* sum2 = stats + 512; float* sq2 = stats + 640;
  float* sc0 = scsh;         float* sh0 = scsh + 128;
  float* sc1 = scsh + 256;   float* sh1 = scsh + 384;
  float* sc2 = scsh + 512;   float* sh2 = scsh + 640;

  const size_t fpsShmem = (size_t)(4 * N_) * sizeof(float) + 8 * sizeof(float)
                        + 8 * sizeof(int) + sizeof(int);

  zero_kernel<<<3, 256, 0, stream>>>(stats, 768);

  fps_kernel<<<B_, 256, fpsShmem, stream>>>(xyz, new_xyz);

  group_kernel<<<(B_ * S_) / 8, 256, 0, stream>>>(xyz, points, new_xyz, feat);

  // layer 1: 67(->68) -> 64, no pre-BN
  mlp_wmma_kernel<64, 68, 67, false><<<(M_ / 16) / (2 * ITERS), 256, 0, stream>>>(
      feat, w0, b0, nullptr, nullptr, act1, sum0, sq0);
  bn_prep_kernel<<<1, 128, 0, stream>>>(sum0, sq0, g0, be0, sc0, sh0, 64);

  // layer 2: 64 -> 64, BN0+ReLU fused into A load
  mlp_wmma_kernel<64, 64, 64, true><<<(M_ / 16) / (2 * ITERS), 256, 0, stream>>>(
      act1, w1, b1, sc0, sh0, act2, sum1, sq1);
  bn_prep_kernel<<<1, 128, 0, stream>>>(sum1, sq1, g1, be1, sc1, sh1, 64);

  // layer 3: 64 -> 128, BN1+ReLU fused into A load
  mlp_wmma_kernel<128, 64, 64, true><<<(M_ / 16) / ITERS, 256, 0, stream>>>(
      act2, w2, b2, sc1, sh1, act3, sum2, sq2);
  bn_prep_kernel<<<1, 128, 0, stream>>>(sum2, sq2, g2, be2, sc2, sh2, 128);

  // BN2 + ReLU + max over K
  maxpool_kernel<<<B_ * S_, 128, 0, stream>>>(act3, sc2, sh2, new_points);
}